// GraphModel_39170101740252
// MI455X (gfx1250) — compile-verified
//
#include <hip/hip_runtime.h>
#include <math.h>

#define N_NODES 30000
#define N_EDGES 480000
#define DD 128
#define EPSF 1e-7f

#define SA_STRIDE 132   // floats per A-panel row in LDS (pad: conflict-free frags)
#define SW_STRIDE 288   // floats per k-pair row of W in LDS (pad: half-waves split)

typedef float v2f __attribute__((ext_vector_type(2)));
typedef float v8f __attribute__((ext_vector_type(8)));

// ---------------------------------------------------------------------------
// Node encoder layer 1: t = relu(x @ Wn1 + bn1), x:[N,7], Wn1:[7,128]
// ---------------------------------------------------------------------------
__global__ __launch_bounds__(256) void enc1_kernel(
    const float* __restrict__ x, const float* __restrict__ Wn1,
    const float* __restrict__ bn1, float* __restrict__ t, int total)
{
    int idx = blockIdx.x * 256 + threadIdx.x;
    if (idx >= total) return;
    int n = idx >> 7;
    int c = idx & 127;
    float s = bn1[c];
#pragma unroll
    for (int j = 0; j < 7; ++j)
        s = fmaf(x[n * 7 + j], Wn1[j * 128 + c], s);
    t[idx] = fmaxf(s, 0.0f);
}

// ---------------------------------------------------------------------------
// WMMA GEMM: C[M,128] = A[M,128] @ W[128,128] + bias[128]
// Block = 8 waves = 128 output rows. A panel staged in LDS via CDNA5 async
// copy (global_load_async_to_lds_b128, ASYNCcnt). The LDS destination address
// is derived from &sA[...] (ptrtoint) so the buffer is captured: the compiler
// must keep the subsequent ds_load fragment reads live, and the offset is
// correct regardless of LDS layout. W staged k-pair-interleaved so each B
// fragment is one contiguous ds_load_b64.
//
// Fragment layouts per CDNA5 ISA 7.12.2 (f32 16x16x4):
//   A 16x4 : lanes0-15 M=lane, v0=K0 v1=K1; lanes16-31 M=lane-16, v0=K2 v1=K3
//   B 4x16 : v0: lanes0-15 K=0 / lanes16-31 K=2 ; v1: K=1 / K=3 ; N=lane&15
//   C 16x16: VGPR v: lanes0-15 M=v, lanes16-31 M=v+8; N=lane&15
// ---------------------------------------------------------------------------
__global__ __launch_bounds__(256) void gemm128_kernel(
    const float* __restrict__ A, const float* __restrict__ W,
    const float* __restrict__ bias, float* __restrict__ C, int M)
{
    __shared__ float sA[128 * SA_STRIDE];   // ~67.6 KB
    __shared__ float sW[64 * SW_STRIDE];    // ~73.7 KB
    int tid = threadIdx.x;
    int rowBlk = blockIdx.x * 128;

    // ---- async global->LDS copy of the A panel (128x128 f32) ----
    {
        const float* Ablk = A + (size_t)rowBlk * 128;
#pragma unroll
        for (int it = 0; it < 16; ++it) {
            int i = tid + it * 256;                       // float4 index 0..4095
            int row = i >> 5;
            int c4  = i & 31;
            // ptrtoint of &sA[..]: captures sA AND yields the wave-relative
            // LDS byte address in the low 32 bits of the flat address.
            unsigned lds_addr =
                (unsigned)(size_t)&sA[row * SA_STRIDE + c4 * 4];
            unsigned g_off = (unsigned)i * 16u;
            asm volatile("global_load_async_to_lds_b128 %0, %1, %2"
                         :: "v"(lds_addr), "v"(g_off), "s"(Ablk)
                         : "memory");
        }
    }

    // ---- synchronous swizzled copy of W: sW[kpair][2*col + (k&1)] ----
#pragma unroll
    for (int it = 0; it < 8; ++it) {
        int task = tid + it * 256;                        // 0..2047
        int r  = task >> 5;                               // k-pair row
        int c4 = task & 31;                               // col group of 4
        float4 w0 = *(const float4*)(W + (size_t)(2 * r) * 128 + c4 * 4);
        float4 w1 = *(const float4*)(W + (size_t)(2 * r + 1) * 128 + c4 * 4);
        v2f* dstp = (v2f*)(sW + r * SW_STRIDE + c4 * 8);
        dstp[0] = (v2f){w0.x, w1.x};
        dstp[1] = (v2f){w0.y, w1.y};
        dstp[2] = (v2f){w0.z, w1.z};
        dstp[3] = (v2f){w0.w, w1.w};
    }

    asm volatile("s_wait_asynccnt 0x0" ::: "memory");
    __syncthreads();

    int wave = tid >> 5;
    int lane = tid & 31;
    int l15  = lane & 15;
    int hi   = lane >> 4;
    int row0 = rowBlk + wave * 16;
    if (row0 >= M) return;        // uniform per wave: EXEC stays all-ones

    const float* ap = sA + (wave * 16 + l15) * SA_STRIDE + 2 * hi;
    const float* bp = sW + hi * SW_STRIDE + l15 * 2;

    v8f acc[8];
#pragma unroll
    for (int n = 0; n < 8; ++n) acc[n] = (v8f){0, 0, 0, 0, 0, 0, 0, 0};

#pragma unroll 4
    for (int k = 0; k < 128; k += 4) {
        v2f a = *(const v2f*)(ap + k);
        const float* bk = bp + (k >> 1) * SW_STRIDE;
#pragma unroll
        for (int n = 0; n < 8; ++n) {
            v2f b = *(const v2f*)(bk + n * 32);
            acc[n] = __builtin_amdgcn_wmma_f32_16x16x4_f32(
                false, a, false, b, (short)0, acc[n], false, false);
        }
    }

#pragma unroll
    for (int n = 0; n < 8; ++n) {
        float bv = bias[n * 16 + l15];
#pragma unroll
        for (int v = 0; v < 8; ++v)
            C[(size_t)(row0 + v + 8 * hi) * 128 + n * 16 + l15] = acc[n][v] + bv;
    }
}

// ---------------------------------------------------------------------------
// Zero the per-layer scatter buffers (m bits / denom / sum(ex*msg))
// ---------------------------------------------------------------------------
__global__ __launch_bounds__(256) void init3_kernel(
    float* __restrict__ m, float* __restrict__ den, float* __restrict__ sxm,
    int total)
{
    int idx = blockIdx.x * 256 + threadIdx.x;
    if (idx >= total) return;
    m[idx] = 0.0f;       // 0 bits: valid identity for positive-float int-max
    den[idx] = 0.0f;
    sxm[idx] = 0.0f;
}

// msg = relu(h[src] + (ea*We + be)) + EPS  (always > 0)
__device__ __forceinline__ float4 edge_msg(const float* __restrict__ H, int s,
                                           float ea,
                                           const float* __restrict__ We,
                                           const float* __restrict__ be, int c)
{
    float4 hv = *(const float4*)(H + (size_t)s * 128 + c);
    float4 wv = *(const float4*)(We + c);
    float4 bv = *(const float4*)(be + c);
    float4 m;
    m.x = fmaxf(hv.x + fmaf(ea, wv.x, bv.x), 0.0f) + EPSF;
    m.y = fmaxf(hv.y + fmaf(ea, wv.y, bv.y), 0.0f) + EPSF;
    m.z = fmaxf(hv.z + fmaf(ea, wv.z, bv.z), 0.0f) + EPSF;
    m.w = fmaxf(hv.w + fmaf(ea, wv.w, bv.w), 0.0f) + EPSF;
    return m;
}

// ---------------------------------------------------------------------------
// Pass 1: segment max of msg into mbits (int atomicMax on positive floats)
// ---------------------------------------------------------------------------
__global__ __launch_bounds__(256) void edge_max_kernel(
    const float* __restrict__ H, const int* __restrict__ src,
    const int* __restrict__ dst, const float* __restrict__ ea,
    const float* __restrict__ We, const float* __restrict__ be,
    int* __restrict__ mbits, int E)
{
    int idx = blockIdx.x * 256 + threadIdx.x;
    int e = idx >> 5;
    if (e >= E) return;
    int lane = idx & 31;
    int s = src[e], d = dst[e];
    float a = ea[e];
    int c = lane * 4;
    float4 m = edge_msg(H, s, a, We, be, c);
    int* p = mbits + (size_t)d * 128 + c;
    atomicMax(p + 0, __float_as_int(m.x));
    atomicMax(p + 1, __float_as_int(m.y));
    atomicMax(p + 2, __float_as_int(m.z));
    atomicMax(p + 3, __float_as_int(m.w));
}

// ---------------------------------------------------------------------------
// Pass 2: denom += exp(msg-m), sxm += exp(msg-m)*msg   (segment sums)
// ---------------------------------------------------------------------------
__global__ __launch_bounds__(256) void edge_sum_kernel(
    const float* __restrict__ H, const int* __restrict__ src,
    const int* __restrict__ dst, const float* __restrict__ ea,
    const float* __restrict__ We, const float* __restrict__ be,
    const float* __restrict__ mbuf, float* __restrict__ den,
    float* __restrict__ sxm, int E)
{
    int idx = blockIdx.x * 256 + threadIdx.x;
    int e = idx >> 5;
    if (e >= E) return;
    int lane = idx & 31;
    int s = src[e], d = dst[e];
    float a = ea[e];
    int c = lane * 4;
    float4 m = edge_msg(H, s, a, We, be, c);
    float4 mx = *(const float4*)(mbuf + (size_t)d * 128 + c);
    float e0 = __expf(m.x - mx.x);
    float e1 = __expf(m.y - mx.y);
    float e2 = __expf(m.z - mx.z);
    float e3 = __expf(m.w - mx.w);
    size_t base = (size_t)d * 128 + c;
    atomicAdd(den + base + 0, e0);
    atomicAdd(den + base + 1, e1);
    atomicAdd(den + base + 2, e2);
    atomicAdd(den + base + 3, e3);
    atomicAdd(sxm + base + 0, e0 * m.x);
    atomicAdd(sxm + base + 1, e1 * m.y);
    atomicAdd(sxm + base + 2, e2 * m.z);
    atomicAdd(sxm + base + 3, e3 * m.w);
}

// ---------------------------------------------------------------------------
// agg = sxm/denom (softmax-weighted sum), residual: t = agg + h
// ---------------------------------------------------------------------------
__global__ __launch_bounds__(256) void finalize_kernel(
    const float* __restrict__ h, const float* __restrict__ den,
    const float* __restrict__ sxm, float* __restrict__ t, int total)
{
    int idx = blockIdx.x * 256 + threadIdx.x;
    if (idx >= total) return;
    float d = den[idx];
    float agg = (d > 0.0f) ? (sxm[idx] / d) : 0.0f;
    t[idx] = agg + h[idx];
}

// ---------------------------------------------------------------------------
// Output head: out[N,3] = h @ Wo + bo.  One wave per node, shuffle reduction.
// ---------------------------------------------------------------------------
__global__ __launch_bounds__(256) void head_kernel(
    const float* __restrict__ H, const float* __restrict__ Wo,
    const float* __restrict__ bo, float* __restrict__ out, int N)
{
    int wave = threadIdx.x >> 5;
    int lane = threadIdx.x & 31;
    int n = blockIdx.x * 8 + wave;
    if (n >= N) return;
    float4 hv = ((const float4*)(H + (size_t)n * 128))[lane];
    float hh[4] = {hv.x, hv.y, hv.z, hv.w};
    int c = lane * 4;
    float p0 = 0.f, p1 = 0.f, p2 = 0.f;
#pragma unroll
    for (int i = 0; i < 4; ++i) {
        const float* w = Wo + (size_t)(c + i) * 3;
        p0 = fmaf(hh[i], w[0], p0);
        p1 = fmaf(hh[i], w[1], p1);
        p2 = fmaf(hh[i], w[2], p2);
    }
#pragma unroll
    for (int off = 16; off > 0; off >>= 1) {
        p0 += __shfl_down(p0, off, 32);
        p1 += __shfl_down(p1, off, 32);
        p2 += __shfl_down(p2, off, 32);
    }
    if (lane == 0) {
        out[(size_t)n * 3 + 0] = p0 + bo[0];
        out[(size_t)n * 3 + 1] = p1 + bo[1];
        out[(size_t)n * 3 + 2] = p2 + bo[2];
    }
}

// ---------------------------------------------------------------------------
extern "C" void kernel_launch(void* const* d_in, const int* in_sizes, int n_in,
                              void* d_out, int out_size, void* d_ws,
                              size_t ws_size, hipStream_t stream)
{
    const float* x   = (const float*)d_in[0];
    const int*   ei  = (const int*)d_in[1];
    const float* ea  = (const float*)d_in[2];
    const float* Wn1 = (const float*)d_in[3];
    const float* bn1 = (const float*)d_in[4];
    const float* Wn2 = (const float*)d_in[5];
    const float* bn2 = (const float*)d_in[6];
    const float* We  = (const float*)d_in[7];
    const float* be  = (const float*)d_in[8];
    const float* Wg  = (const float*)d_in[9];
    const float* bg  = (const float*)d_in[10];
    const float* Wo  = (const float*)d_in[11];
    const float* bo  = (const float*)d_in[12];
    float* out = (float*)d_out;

    const int N = N_NODES, E = N_EDGES;
    const size_t ND = (size_t)N * DD;

    float* ws    = (float*)d_ws;
    float* h     = ws;            // [N,128]
    float* t     = ws + ND;       // [N,128]
    float* mbuf  = ws + 2 * ND;   // [N,128] segment max (int bits)
    float* den   = ws + 3 * ND;   // [N,128] sum exp
    float* sxm   = ws + 4 * ND;   // [N,128] sum exp*msg

    const int* src = ei;
    const int* dst = ei + E;

    const int ndBlocks  = (int)((ND + 255) / 256);          // 15000
    const int gemmBlks  = (N + 127) / 128;                  // 235
    const int edgeBlks  = (E * 32 + 255) / 256;             // 60000
    const int headBlks  = (N + 7) / 8;                      // 3750

    // Node encoder
    enc1_kernel<<<ndBlocks, 256, 0, stream>>>(x, Wn1, bn1, t, (int)ND);
    gemm128_kernel<<<gemmBlks, 256, 0, stream>>>(t, Wn2, bn2, h, N);

    // 4 GENConv layers
    for (int l = 0; l < 4; ++l) {
        init3_kernel<<<ndBlocks, 256, 0, stream>>>(mbuf, den, sxm, (int)ND);
        edge_max_kernel<<<edgeBlks, 256, 0, stream>>>(h, src, dst, ea, We, be,
                                                      (int*)mbuf, E);
        edge_sum_kernel<<<edgeBlks, 256, 0, stream>>>(h, src, dst, ea, We, be,
                                                      mbuf, den, sxm, E);
        finalize_kernel<<<ndBlocks, 256, 0, stream>>>(h, den, sxm, t, (int)ND);
        gemm128_kernel<<<gemmBlks, 256, 0, stream>>>(
            t, Wg + (size_t)l * DD * DD, bg + (size_t)l * DD, h, N);
    }

    // Output head
    head_kernel<<<headBlks, 256, 0, stream>>>(h, Wo, bo, out, N);
}